// BillehColumnTorch_83236466196564
// MI455X (gfx1250) — compile-verified
//
#include <hip/hip_runtime.h>
#include <cstdint>

#ifndef __has_builtin
#define __has_builtin(x) 0
#endif
#if __has_builtin(__builtin_amdgcn_tensor_load_to_lds) && __has_builtin(__builtin_amdgcn_s_wait_tensorcnt)
#define HAS_TDM 1
#else
#define HAS_TDM 0
#endif

// ---------------- problem constants (match reference) ----------------
constexpr int   kN    = 50000;
constexpr int   kR    = 4;
constexpr int   kD    = 5;
constexpr int   kNIN  = 17400;
constexpr int   kB    = 2;
constexpr int   kT    = 50;
constexpr int   kEREC = 2000000;
constexpr int   kEIN  = 500000;
constexpr int   kRN   = kR * kN;
constexpr float kDT   = 1.0f;

// ---------------- TDM staging config ----------------
constexpr int CHUNK  = 2048;          // edges per LDS chunk
constexpr int SLICE  = CHUNK / 8;     // per-wave slice (8 waves of 32)
constexpr int kNCH   = (kEREC + CHUNK - 1) / CHUNK;
constexpr int kPadE  = kNCH * CHUNK;  // padded edge count

// ---------------- workspace layout (float units) ----------------
constexpr size_t OFF_CUR  = 0;         // B*R*N   = 400000
constexpr size_t OFF_V    = 400000;    // B*N
constexpr size_t OFF_RF   = 500000;    // B*N
constexpr size_t OFF_A1   = 600000;    // B*N
constexpr size_t OFF_A2   = 700000;    // B*N
constexpr size_t OFF_PR   = 800000;    // B*N*R
constexpr size_t OFF_PS   = 1200000;   // B*N*R
constexpr size_t OFF_Z    = 1600000;   // B*D*N
constexpr size_t OFF_AD   = 2100000;   // N*2 (asc_decay)
constexpr size_t OFF_EDGE = 2200000;   // padded rec edges: int rows, int cols, float w
constexpr size_t STATE_FLOATS = OFF_EDGE;

typedef unsigned int u32x4 __attribute__((ext_vector_type(4)));
typedef int          i32x4 __attribute__((ext_vector_type(4)));
typedef int          i32x8 __attribute__((ext_vector_type(8)));

#if HAS_TDM
// Build a D# for a 1-D contiguous load of `len` dwords from `gptr` into LDS
// offset `lds_off` and issue TENSOR_LOAD_TO_LDS. (data_size=4B; tile_dim0=len;
// higher dims degenerate to 1.)
__device__ __forceinline__ void tdm_load_1d(unsigned lds_off, const void* gptr, unsigned len) {
  unsigned long long ga = (unsigned long long)(uintptr_t)gptr;
  u32x4 g0;
  g0[0] = 1u;                                               // count=1 (valid user descriptor)
  g0[1] = lds_off;                                          // lds_addr
  g0[2] = (unsigned)ga;                                     // global_addr[31:0]
  g0[3] = (unsigned)((ga >> 32) & 0x01FFFFFFull) | (2u << 30); // addr[56:32] | type=2
  i32x8 g1;
  g1[0] = (int)(2u << 16);                                  // data_size=2 (4 bytes), no mask
  g1[1] = (int)((len & 0xFFFFu) << 16);                     // tensor_dim0 lo16
  g1[2] = (int)(((len >> 16) & 0xFFFFu) | (1u << 16));      // tensor_dim0 hi16 | tensor_dim1=1 lo16
  g1[3] = (int)((len & 0xFFFFu) << 16);                     // tensor_dim1 hi16=0 | tile_dim0=len
  g1[4] = 1;                                                // tile_dim1=1, tile_dim2=0
  g1[5] = (int)len;                                         // tensor_dim0_stride lo32
  g1[6] = (int)((len & 0xFFFFu) << 16);                     // stride0 hi16=0 | stride1 lo16
  g1[7] = (int)(len >> 16);                                 // stride1 hi32
  i32x4 g2; g2[0] = 1; g2[1] = 1; g2[2] = (int)len; g2[3] = 0;        // dim2=1, dim3=1, stride2
  i32x4 g3; g3[0] = (int)len; g3[1] = (1 << 16); g3[2] = 0; g3[3] = 0; // stride3, dim4=1, tile4=0
  i32x8 g4 = {0, 0, 0, 0, 0, 0, 0, 0};                      // extra group (clang-23 6-arg form)
  __builtin_amdgcn_tensor_load_to_lds(g0, g1, g2, g3, g4, 0);
}
#endif

// ---------------- init: state, asc_decay, padded edge copy ----------------
__global__ __launch_bounds__(256) void init_kernel(
    float* __restrict__ ws, const float* __restrict__ v_reset,
    const float* __restrict__ param_k,
    const int* __restrict__ rrows, const int* __restrict__ rcols,
    const float* __restrict__ rw, int pad_e) {
  int i0 = blockIdx.x * blockDim.x + threadIdx.x;
  int stride = gridDim.x * blockDim.x;
  float* v  = ws + OFF_V;
  float* ad = ws + OFF_AD;
  int*   prow = (int*)(ws + OFF_EDGE);
  int*   pcol = prow + kPadE;
  float* pw   = (float*)(pcol + kPadE);
  for (int i = i0; i < kB * kN; i += stride) v[i] = v_reset[i % kN];
  for (int i = i0; i < 2 * kN; i += stride) ad[i] = expf(-kDT * param_k[i]);
  for (int i = i0; i < pad_e; i += stride) {
    bool ok = i < kEREC;
    prow[i] = ok ? rrows[i] : 0;
    pcol[i] = ok ? rcols[i] : 0;
    pw[i]   = ok ? rw[i]    : 0.0f;
  }
}

// ---------------- input scatter: cur[b][row] += x[b,t,col] * w ----------------
__global__ __launch_bounds__(256) void in_scatter(
    const int* __restrict__ rows, const int* __restrict__ cols,
    const float* __restrict__ w, const float* __restrict__ x,
    float* __restrict__ cur, int t) {
  int e = blockIdx.x * blockDim.x + threadIdx.x;
  if (e >= kEIN) return;
  int row = rows[e], col = cols[e];
  float wt = w[e];
  float x0 = x[(0 * kT + t) * kNIN + col];
  float x1 = x[(1 * kT + t) * kNIN + col];
  unsafeAtomicAdd(cur + row, x0 * wt);
  unsafeAtomicAdd(cur + kRN + row, x1 * wt);
}

// ---------------- recurrent scatter, TDM double-buffered ----------------
#if HAS_TDM
__global__ __launch_bounds__(256) void rec_scatter_tdm(
    const int* __restrict__ rows, const int* __restrict__ cols,
    const float* __restrict__ w, const float* __restrict__ zbuf,
    float* __restrict__ cur, int nch, int head) {
  __shared__ int   lrow[2][CHUNK];
  __shared__ int   lcol[2][CHUNK];
  __shared__ float lw[2][CHUNK];
  const int tid  = threadIdx.x;
  const int wave = tid >> 5;
  const unsigned so = (unsigned)(wave * SLICE);   // wave-uniform slice offset

  int c = blockIdx.x;
  if (c < nch) {  // prime buffer 0 (each wave DMAs its own slice of all 3 arrays)
    size_t base = (size_t)c * CHUNK + so;
    tdm_load_1d((unsigned)(uintptr_t)&lrow[0][so], rows + base, SLICE);
    tdm_load_1d((unsigned)(uintptr_t)&lcol[0][so], cols + base, SLICE);
    tdm_load_1d((unsigned)(uintptr_t)&lw[0][so],   w    + base, SLICE);
  }
  int buf = 0;
  while (c < nch) {
    int cn = c + (int)gridDim.x;
    if (cn < nch) {  // prefetch next chunk into the other buffer, then wait current
      size_t base = (size_t)cn * CHUNK + so;
      tdm_load_1d((unsigned)(uintptr_t)&lrow[buf ^ 1][so], rows + base, SLICE);
      tdm_load_1d((unsigned)(uintptr_t)&lcol[buf ^ 1][so], cols + base, SLICE);
      tdm_load_1d((unsigned)(uintptr_t)&lw[buf ^ 1][so],   w    + base, SLICE);
      __builtin_amdgcn_s_wait_tensorcnt(3);   // current chunk's 3 loads done (in-order)
    } else {
      __builtin_amdgcn_s_wait_tensorcnt(0);
    }
    __syncthreads();
#pragma unroll
    for (int i = 0; i < CHUNK / 256; ++i) {
      int e = tid + (i << 8);
      float wt = lw[buf][e];
      if (wt != 0.0f) {                   // padding has w==0
        int col = lcol[buf][e];
        int d = col / kN;                 // logical delay slot
        int n = col - d * kN;
        int slot = head + d; if (slot >= kD) slot -= kD;   // ring buffer
        int row = lrow[buf][e];
        float z0 = zbuf[slot * kN + n];           // b=0
        float z1 = zbuf[(kD + slot) * kN + n];    // b=1
        if (z0 != 0.0f) unsafeAtomicAdd(cur + row, wt);          // z is exactly 1
        if (z1 != 0.0f) unsafeAtomicAdd(cur + kRN + row, wt);
      }
    }
    __syncthreads();   // all reads of `buf` done before it becomes a prefetch target
    buf ^= 1;
    c = cn;
  }
}
#endif

// ---------------- recurrent scatter, plain fallback ----------------
__global__ __launch_bounds__(256) void rec_scatter_plain(
    const int* __restrict__ rows, const int* __restrict__ cols,
    const float* __restrict__ w, const float* __restrict__ zbuf,
    float* __restrict__ cur, int head) {
  int e = blockIdx.x * blockDim.x + threadIdx.x;
  if (e >= kEREC) return;
  int col = cols[e];
  float wt = w[e];
  int d = col / kN;
  int n = col - d * kN;
  int slot = head + d; if (slot >= kD) slot -= kD;
  int row = rows[e];
  float z0 = zbuf[slot * kN + n];
  float z1 = zbuf[(kD + slot) * kN + n];
  if (z0 != 0.0f) unsafeAtomicAdd(cur + row, wt);
  if (z1 != 0.0f) unsafeAtomicAdd(cur + kRN + row, wt);
}

// ---------------- fused per-neuron GLIF3 update ----------------
__global__ __launch_bounds__(256) void neuron_step(
    float* __restrict__ ws,
    const float* __restrict__ bkg,   const float* __restrict__ v_th,
    const float* __restrict__ e_l,   const float* __restrict__ v_reset,
    const float* __restrict__ g,     const float* __restrict__ t_ref,
    const float* __restrict__ asc_amps, const float* __restrict__ decay,
    const float* __restrict__ cf,    const float* __restrict__ syn_decay,
    const float* __restrict__ psc_init, const float* __restrict__ vscale,
    const float* __restrict__ voff,  const int* __restrict__ rest,
    float* __restrict__ outz, float* __restrict__ outv,
    int t, int head, int new_head) {
  int i = blockIdx.x * blockDim.x + threadIdx.x;
  if (i >= kB * kN) return;
  int b = i / kN, n = i - b * kN;

  float4* cur4 = (float4*)(ws + OFF_CUR);
  float*  V    = ws + OFF_V;
  float*  RF   = ws + OFF_RF;
  float*  A1   = ws + OFF_A1;
  float*  A2   = ws + OFF_A2;
  float4* pr4  = (float4*)(ws + OFF_PR);
  float4* ps4  = (float4*)(ws + OFF_PS);
  float*  zbuf = ws + OFF_Z;
  const float2* ad2  = (const float2*)(ws + OFF_AD);
  const float4* bkg4 = (const float4*)bkg;
  const float4* sd4  = (const float4*)syn_decay;
  const float4* pi4  = (const float4*)psc_init;
  const float2* am2  = (const float2*)asc_amps;

  float rest_sc = (float)rest[t * kB + b] * 0.1f;
  float prev_z  = zbuf[(b * kD + head) * kN + n];

  float vv = V[i], rr = RF[i], aa1 = A1[i], aa2 = A2[i];
  float4 p  = ps4[i];
  float4 pr = pr4[i];
  float4 rin = cur4[i];
  float4 bk  = bkg4[n];
  rin.x += bk.x * rest_sc; rin.y += bk.y * rest_sc;
  rin.z += bk.z * rest_sc; rin.w += bk.w * rest_sc;

  float vth = v_th[n], el = e_l[n];
  float input_current = p.x + p.y + p.z + p.w;          // OLD psc
  float leak = g[n] * el;
  float new_v = decay[n] * vv
              + cf[n] * (input_current + aa1 + aa2 + leak)   // OLD a1,a2
              + prev_z * (v_reset[n] - vth);

  float4 sd = sd4[n], pi = pi4[n];
  float4 np, npr;
  np.x = p.x * sd.x + kDT * sd.x * pr.x;  npr.x = sd.x * pr.x + rin.x * pi.x;
  np.y = p.y * sd.y + kDT * sd.y * pr.y;  npr.y = sd.y * pr.y + rin.y * pi.y;
  np.z = p.z * sd.z + kDT * sd.z * pr.z;  npr.z = sd.z * pr.z + rin.z * pi.z;
  np.w = p.w * sd.w + kDT * sd.w * pr.w;  npr.w = sd.w * pr.w + rin.w * pi.w;

  float new_r = fmaxf(0.0f, rr + prev_z * t_ref[n] - kDT);
  float2 ad = ad2[n], am = am2[n];
  float na1 = ad.x * aa1 + prev_z * am.x;
  float na2 = ad.y * aa2 + prev_z * am.y;

  float v_sc = (new_v - vth) / (vth - el);
  float nz = (v_sc > 0.0f) ? 1.0f : 0.0f;
  if (new_r > 0.0f) nz = 0.0f;

  zbuf[(b * kD + new_head) * kN + n] = nz;
  outz[(b * kT + t) * kN + n] = nz;
  outv[(b * kT + t) * kN + n] = new_v * vscale[n] + voff[n];

  V[i] = new_v; RF[i] = new_r; A1[i] = na1; A2[i] = na2;
  ps4[i] = np; pr4[i] = npr;
  cur4[i] = make_float4(0.f, 0.f, 0.f, 0.f);   // accumulator ready for next step
}

// ---------------- host-side orchestration ----------------
extern "C" void kernel_launch(void* const* d_in, const int* in_sizes, int n_in,
                              void* d_out, int out_size, void* d_ws, size_t ws_size,
                              hipStream_t stream) {
  const float* x        = (const float*)d_in[0];
  const float* rec_w    = (const float*)d_in[1];
  const float* in_w     = (const float*)d_in[2];
  const float* bkg      = (const float*)d_in[3];
  const float* v_th     = (const float*)d_in[4];
  const float* e_l      = (const float*)d_in[5];
  const float* v_reset  = (const float*)d_in[6];
  const float* g        = (const float*)d_in[7];
  const float* t_ref    = (const float*)d_in[8];
  const float* asc_amps = (const float*)d_in[9];
  const float* param_k  = (const float*)d_in[10];
  const float* decay    = (const float*)d_in[11];
  const float* cf       = (const float*)d_in[12];
  const float* syn_dec  = (const float*)d_in[13];
  const float* psc_init = (const float*)d_in[14];
  const float* vscale   = (const float*)d_in[15];
  const float* voff     = (const float*)d_in[16];
  const int*   rec_rows = (const int*)d_in[17];
  const int*   rec_cols = (const int*)d_in[18];
  const int*   in_rows  = (const int*)d_in[19];
  const int*   in_cols  = (const int*)d_in[20];
  const int*   rest     = (const int*)d_in[21];

  float* ws   = (float*)d_ws;
  float* cur  = ws + OFF_CUR;
  float* zbuf = ws + OFF_Z;
  int*   prow = (int*)(ws + OFF_EDGE);
  int*   pcol = prow + kPadE;
  float* pw   = (float*)(pcol + kPadE);

  size_t needed = (OFF_EDGE + 3 * (size_t)kPadE) * sizeof(float);
  bool use_tdm = (HAS_TDM != 0) && (ws_size >= needed);

  float* outz = (float*)d_out;
  float* outv = outz + (size_t)kB * kT * kN;

  // zero all dynamic state (cur, v, r, a1, a2, psc_rise, psc, zbuf, asc_decay)
  (void)hipMemsetAsync(ws, 0, STATE_FLOATS * sizeof(float), stream);
  init_kernel<<<2048, 256, 0, stream>>>(ws, v_reset, param_k, rec_rows, rec_cols,
                                        rec_w, use_tdm ? kPadE : 0);

  const int gIn  = (kEIN + 255) / 256;
  const int gRec = (kEREC + 255) / 256;
  const int gNeu = (kB * kN + 255) / 256;

  int head = 0;
  for (int t = 0; t < kT; ++t) {
    in_scatter<<<gIn, 256, 0, stream>>>(in_rows, in_cols, in_w, x, cur, t);
#if HAS_TDM
    if (use_tdm) {
      rec_scatter_tdm<<<256, 256, 0, stream>>>(prow, pcol, pw, zbuf, cur, kNCH, head);
    } else
#endif
    {
      rec_scatter_plain<<<gRec, 256, 0, stream>>>(rec_rows, rec_cols, rec_w, zbuf, cur, head);
    }
    int new_head = (head + kD - 1) % kD;
    neuron_step<<<gNeu, 256, 0, stream>>>(ws, bkg, v_th, e_l, v_reset, g, t_ref,
                                          asc_amps, decay, cf, syn_dec, psc_init,
                                          vscale, voff, rest, outz, outv,
                                          t, head, new_head);
    head = new_head;
  }
}